// Model_9466107920822
// MI455X (gfx1250) — compile-verified
//
#include <hip/hip_runtime.h>
#include <hip/hip_bf16.h>

typedef __attribute__((ext_vector_type(2))) float v2f;
typedef __attribute__((ext_vector_type(8))) float v8f;

#define H 64
#define B 16
#define L 512
#define OUT 2
#define HH (H * H)            // 4096 floats = 16 KB per token matrix
#define CHUNKS 32             // workgroups per batch
#define TOK 32                // tokens per workgroup (2*L / CHUNKS)

// ---------------------------------------------------------------------------
// Kernel 0: zero the per-batch accumulators in workspace (required every call:
// harness does not re-zero d_ws between graph replays).
// ---------------------------------------------------------------------------
__global__ __launch_bounds__(256) void zero_ws(float* ws, int n) {
    int i = blockIdx.x * blockDim.x + threadIdx.x;
    if (i < n) ws[i] = 0.0f;
}

// ---------------------------------------------------------------------------
// Kernel 1: streaming accumulation (the 256 MB, bandwidth-bound phase).
// grid = (CHUNKS, B), block = 256 threads (8 wave32s).
// Each thread owns 16 f32 accumulator lanes (4 x float4), reads each token's
// 16 KB matrix with 4 coalesced b128 loads, double-buffered across tokens,
// then commits once with 16 global f32 atomics.
// ---------------------------------------------------------------------------
__global__ __launch_bounds__(256) void accum_tokens(const int* __restrict__ s1,
                                                    const int* __restrict__ s2,
                                                    const float* __restrict__ embed,
                                                    float* __restrict__ ws) {
    const int b   = blockIdx.y;           // batch
    const int t0  = blockIdx.x * TOK;     // first logical token (0..1023)
    const int tid = threadIdx.x;

    auto tok_base = [&](int t) -> const float4* {
        int idx = (t < L) ? s1[b * L + t] : s2[b * L + (t - L)];
        return (const float4*)(embed + (size_t)idx * HH);
    };

    float4 acc[4];
    #pragma unroll
    for (int s = 0; s < 4; ++s) acc[s] = make_float4(0.f, 0.f, 0.f, 0.f);

    // prime double buffer
    float4 c[4];
    {
        const float4* p = tok_base(t0);
        #pragma unroll
        for (int s = 0; s < 4; ++s) c[s] = p[tid + s * 256];
    }

    for (int t = t0; t < t0 + TOK; ++t) {
        float4 n[4];
        #pragma unroll
        for (int s = 0; s < 4; ++s) n[s] = c[s];
        if (t + 1 < t0 + TOK) {
            const float4* q = tok_base(t + 1);
            #pragma unroll
            for (int s = 0; s < 4; ++s) n[s] = q[tid + s * 256];
        }
        if (t + 2 < t0 + TOK) {
            // CDNA5 global_prefetch_b8 of the token-after-next (one line/lane)
            __builtin_prefetch((const void*)(tok_base(t + 2) + tid), 0, 1);
        }
        #pragma unroll
        for (int s = 0; s < 4; ++s) {
            acc[s].x += c[s].x; acc[s].y += c[s].y;
            acc[s].z += c[s].z; acc[s].w += c[s].w;
        }
        #pragma unroll
        for (int s = 0; s < 4; ++s) c[s] = n[s];
    }

    float* dst = ws + (size_t)b * HH;
    #pragma unroll
    for (int s = 0; s < 4; ++s) {
        int e = (s * 256 + tid) * 4;
        atomicAdd(&dst[e + 0], acc[s].x);
        atomicAdd(&dst[e + 1], acc[s].y);
        atomicAdd(&dst[e + 2], acc[s].z);
        atomicAdd(&dst[e + 3], acc[s].w);
    }
}

// ---------------------------------------------------------------------------
// Kernel 2: head. One block of 512 threads (16 wave32s).
//  (a) waves 0..15: h[b,:] = sos^T * Msum_b          (VALU, trivial)
//  (b) waves 0..3:  x = relu(h @ w1^T + b1)           (V_WMMA_F32_16X16X4_F32)
//  (c) lanes 0..31: out = x @ w2^T + b2               (VALU, 32 outputs)
// ---------------------------------------------------------------------------
__global__ __launch_bounds__(512) void head_mlp(const float* __restrict__ ws,
                                                const float* __restrict__ sos,
                                                const float* __restrict__ w1,
                                                const float* __restrict__ b1,
                                                const float* __restrict__ w2,
                                                const float* __restrict__ b2,
                                                float* __restrict__ out) {
    __shared__ float hbuf[B * H];   // h[16][64]
    __shared__ float xbuf[B * H];   // relu(h@w1^T+b1)[16][64]

    const int tid  = threadIdx.x;
    const int wave = tid >> 5;
    const int lane = tid & 31;

    // (a) sos contraction; wave == batch, each lane covers columns lane, lane+32
    if (wave < B) {
        const float* M = ws + (size_t)wave * HH;
        float a0 = 0.f, a1 = 0.f;
        #pragma unroll 4
        for (int w = 0; w < H; ++w) {
            float s = sos[w];
            a0 += s * M[w * H + lane];
            a1 += s * M[w * H + 32 + lane];
        }
        hbuf[wave * H + lane]      = a0;
        hbuf[wave * H + 32 + lane] = a1;
    }
    __syncthreads();

    // (b) GEMM [16x64] = [16x64]x[64x64]^T with f32 WMMA, one 16-wide N-tile per wave
    if (wave < 4) {
        const int n0    = wave * 16;
        const int mn    = lane & 15;            // M index for A rows / N index for B cols
        const int khalf = (lane >> 4) * 2;      // lanes 16..31 carry K=2,3
#if defined(__HIP_DEVICE_COMPILE__)
        v8f acc = {};
        #pragma unroll
        for (int k0 = 0; k0 < H; k0 += 4) {
            // A (16x4 f32): vgpr0 = A[m, k0+khalf], vgpr1 = A[m, k0+khalf+1]
            v2f a = *(const v2f*)&hbuf[mn * H + k0 + khalf];
            // B (4x16 f32): B[k, n] = w1[n, k]  (torch Linear: x @ W^T)
            v2f bv = *(const v2f*)&w1[(n0 + mn) * H + k0 + khalf];
            acc = __builtin_amdgcn_wmma_f32_16x16x4_f32(
                /*neg_a=*/false, a, /*neg_b=*/false, bv,
                /*c_mod=*/(short)0, acc, /*reuse_a=*/false, /*reuse_b=*/false);
        }
        // C/D 16x16 f32 layout: vgpr r, lane l -> M = r + 8*(l>=16), N = l&15
        #pragma unroll
        for (int r = 0; r < 8; ++r) {
            int m = r + 8 * (lane >> 4);
            int n = n0 + (lane & 15);
            float v = acc[r] + b1[n];
            xbuf[m * H + n] = v > 0.f ? v : 0.f;
        }
#else
        // host-pass placeholder (never executed on device): scalar equivalent
        (void)khalf;
        for (int m = mn; m < B; m += 32) {
            for (int n = n0; n < n0 + 16; ++n) {
                float v = b1[n];
                for (int k = 0; k < H; ++k) v += hbuf[m * H + k] * w1[n * H + k];
                xbuf[m * H + n] = v > 0.f ? v : 0.f;
            }
        }
#endif
    }
    __syncthreads();

    // (c) final 64 -> 2 layer: 32 outputs, one per lane of wave 0
    if (tid < B * OUT) {
        int bb = tid & 15, o = tid >> 4;
        float s = b2[o];
        #pragma unroll 4
        for (int k = 0; k < H; ++k) s += xbuf[bb * H + k] * w2[o * H + k];
        out[bb * OUT + o] = s;
    }
}

// ---------------------------------------------------------------------------
extern "C" void kernel_launch(void* const* d_in, const int* in_sizes, int n_in,
                              void* d_out, int out_size, void* d_ws, size_t ws_size,
                              hipStream_t stream) {
    const int*   s1    = (const int*)d_in[0];
    const int*   s2    = (const int*)d_in[1];
    const float* embed = (const float*)d_in[2];
    const float* sos   = (const float*)d_in[3];
    const float* w1    = (const float*)d_in[4];
    const float* b1    = (const float*)d_in[5];
    const float* w2    = (const float*)d_in[6];
    const float* b2    = (const float*)d_in[7];
    float*       out   = (float*)d_out;
    float*       ws    = (float*)d_ws;          // B * H * H floats = 256 KB

    const int nacc = B * HH;
    zero_ws<<<(nacc + 255) / 256, 256, 0, stream>>>(ws, nacc);
    accum_tokens<<<dim3(CHUNKS, B), 256, 0, stream>>>(s1, s2, embed, ws);
    head_mlp<<<1, 512, 0, stream>>>(ws, sos, w1, b1, w2, b2, out);
}